// EnhancedVariancePooling_8091718386010
// MI455X (gfx1250) — compile-verified
//
#include <hip/hip_runtime.h>
#include <cstdint>

// EnhancedVariancePooling: sliding-window variance (K=75, S=15) -> clip -> log
//   x: (64,128,4000) f32   out: (64,128,262) f32
//
// Memory-bound: ~140 MB minimum traffic -> ~6us at 23.3 TB/s. Strategy:
// one workgroup per row; TDM (tensor_load_to_lds) DMAs the 16 KB row into
// LDS (no VGPR staging), then segment-decomposed variance (each window =
// sum of 5 stride-15 segments) so global memory is read exactly once.

#define KSIZE   75
#define WSTRIDE 15
#define T_LEN   4000
#define OUT_LEN 262               // (4000-75)/15 + 1
#define NSEG    266               // OUT_LEN + KSIZE/WSTRIDE - 1
#define NROWS   (64*128)
#define BLOCK   288               // 9 wave32s; covers NSEG and OUT_LEN

typedef unsigned int u32;
typedef __attribute__((ext_vector_type(4))) unsigned int v4u;
typedef __attribute__((ext_vector_type(4))) int          v4i;
typedef __attribute__((ext_vector_type(8))) int          v8i;

__global__ __launch_bounds__(BLOCK)
void var_pool_kernel(const float* __restrict__ x, float* __restrict__ out) {
    __shared__ float lds_x[T_LEN];
    __shared__ float lds_S[NSEG];
    __shared__ float lds_Q[NSEG];

    const int row = blockIdx.x;
    const int tid = threadIdx.x;

    // ---- Stage 1: DMA one row (16000 B) global -> LDS via Tensor Data Mover ----
#if defined(__has_builtin) && __has_builtin(__builtin_amdgcn_tensor_load_to_lds)
    if (tid < 32) {  // wave 0 issues the TDM op (SGPR descriptor, EXEC ignored)
        // LDS byte offset of lds_x within this workgroup's LDS allocation
        unsigned lds_off =
            (unsigned)(uintptr_t)(__attribute__((address_space(3))) float*)lds_x;
        unsigned long long ga =
            (unsigned long long)(uintptr_t)(x + (size_t)row * T_LEN);

        // D# group 0 (ISA 8.3): count=1, lds_addr, global_addr[56:0], type=2
        v4u g0;
        g0.x = 1u;                                          // count=1, user mode
        g0.y = lds_off;                                     // lds_addr (bytes)
        g0.z = (u32)(ga & 0xFFFFFFFFu);                     // global_addr[31:0]
        g0.w = (u32)((ga >> 32) & 0x1FFFFFFu) | (2u << 30); // addr[56:32] | type=2

        // D# group 1 (ISA 8.4): 1-row tile of 4000 4-byte elements
        v8i g1;
        g1[0] = (int)(2u << 16);                            // data_size=2 -> 4B; mask=0
        g1[1] = (int)(((u32)T_LEN & 0xFFFFu) << 16);        // tensor_dim0[15:0]
        g1[2] = (int)((((u32)T_LEN >> 16) & 0xFFFFu)        // tensor_dim0[31:16]
                      | (1u << 16));                        // tensor_dim1 = 1
        g1[3] = (int)(((u32)T_LEN & 0xFFFFu) << 16);        // tile_dim0 = 4000
        g1[4] = 0;                                          // tile_dim1/2 unused
        g1[5] = (int)T_LEN;                                 // tensor_dim0_stride
        g1[6] = 0;                                          // stride hi / dim1 stride
        g1[7] = 0;
        v4i gz = {0, 0, 0, 0};
#if __clang_major__ >= 23
        v8i gz8 = {0, 0, 0, 0, 0, 0, 0, 0};
        __builtin_amdgcn_tensor_load_to_lds(g0, g1, gz, gz, gz8, 0);
#else
        __builtin_amdgcn_tensor_load_to_lds(g0, g1, gz, gz, 0);
#endif
        __builtin_amdgcn_s_wait_tensorcnt(0);
    }
    __syncthreads();
#else
    // Fallback: plain cooperative copy
    for (int i = tid; i < T_LEN; i += BLOCK)
        lds_x[i] = x[(size_t)row * T_LEN + i];
    __syncthreads();
#endif

    // ---- Stage 2: per-segment partials (15 elems -> sum, sumsq) ----
    // Thread j reads lds_x[j*15 + k]: stride-15 across lanes, gcd(15,64)=1
    // -> bank-conflict-free within a wave32.
    if (tid < NSEG) {
        const float* p = &lds_x[tid * WSTRIDE];
        float s = 0.0f, q = 0.0f;
#pragma unroll
        for (int k = 0; k < WSTRIDE; ++k) {
            float v = p[k];
            s += v;
            q += v * v;
        }
        lds_S[tid] = s;
        lds_Q[tid] = q;
    }
    __syncthreads();

    // ---- Stage 3: window = 5 consecutive segments; var(ddof=1) -> clip -> log ----
    if (tid < OUT_LEN) {
        float s = 0.0f, q = 0.0f;
#pragma unroll
        for (int k = 0; k < 5; ++k) {
            s += lds_S[tid + k];
            q += lds_Q[tid + k];
        }
        const float n = (float)KSIZE;
        float var = (q - s * s / n) / (n - 1.0f);
        var = fminf(fmaxf(var, 1e-6f), 1e6f);
        out[(size_t)row * OUT_LEN + tid] = logf(var);
    }
}

extern "C" void kernel_launch(void* const* d_in, const int* in_sizes, int n_in,
                              void* d_out, int out_size, void* d_ws, size_t ws_size,
                              hipStream_t stream) {
    const float* x = (const float*)d_in[0];
    float* out = (float*)d_out;
    (void)in_sizes; (void)n_in; (void)out_size; (void)d_ws; (void)ws_size;
    var_pool_kernel<<<NROWS, BLOCK, 0, stream>>>(x, out);
}